// RnnAttentionDecoder_13967233647631
// MI455X (gfx1250) — compile-verified
//
#include <hip/hip_runtime.h>
#include <math.h>

// ---------------------------------------------------------------------------
// Problem constants (match the JAX reference)
// ---------------------------------------------------------------------------
#define Bc   256   // batch
#define Sc   256   // source length
#define Dc   512   // encoder feature dim
#define Hc   512   // hidden
#define Ec   256   // embedding dim
#define OUTc 101   // generator classes (V+1)
#define DEc  768   // D + E
#define G4c  2048  // 4*H
#define SOSc 101   // NEMB-1

typedef __attribute__((ext_vector_type(2))) float v2f;
typedef __attribute__((ext_vector_type(8))) float v8f;

// ---------------------------------------------------------------------------
// Generic fp32 WMMA GEMM:  C[M,N] = A[M,K] @ W[N,K]^T (+ bias[N]) (+= C)
// V_WMMA_F32_16X16X4_F32; one wave -> 16 x (16*NB) strip.
// K processed in 16-wide chunks, register double-buffered so chunk k+1's
// global loads overlap chunk k's WMMAs. Tile->wave mapping is M-fastest so
// the 8 waves of a block share the same W (B-operand) addresses, which
// dedupe via WGP$ broadcast (B is 4x the A traffic).
// Requires K % 32 == 0 (true for 512 and 768 here).
// ---------------------------------------------------------------------------
template <int NB, bool ACCUM, bool NGUARD>
__global__ __launch_bounds__(256) void gemm_nt_wmma(
    const float* __restrict__ A, const float* __restrict__ W,
    const float* __restrict__ bias, float* __restrict__ C,
    int M, int Npad, int K, int Nreal, int ldc)
{
  const int lane  = threadIdx.x & 31;
  const int wave  = threadIdx.x >> 5;
  const int tilesN = Npad / (16 * NB);
  const int tilesM = M >> 4;
  const int tile = blockIdx.x * 8 + wave;
  if (tile >= tilesM * tilesN) return;            // wave-uniform exit

  const int tm = tile % tilesM;                   // M-fastest: block shares tn
  const int tn = tile / tilesM;
  const int m0 = tm << 4;
  const int n0 = tn * (16 * NB);
  const int r15 = lane & 15;
  const int kh  = (lane >> 4) << 1;               // 0 or 2

  const float* __restrict__ Arow = A + (size_t)(m0 + r15) * (size_t)K + kh;
  const float* Wrow[NB];
  bool nok[NB];
#pragma unroll
  for (int j = 0; j < NB; ++j) {
    const int n = n0 + j * 16 + r15;
    nok[j] = (!NGUARD) || (n < Nreal);
    Wrow[j] = W + (size_t)(nok[j] ? n : 0) * (size_t)K + kh;
  }

  v8f acc[NB] = {};

  // fragment chunks: 4 k-steps each, two buffers
  v2f a0[4], b0[4][NB];
  v2f a1[4], b1[4][NB];

#define LOAD_CHUNK(abuf, bbuf, kk)                                      \
  do {                                                                  \
    _Pragma("unroll")                                                   \
    for (int i = 0; i < 4; ++i) {                                       \
      abuf[i].x = Arow[(kk) + 4 * i];                                   \
      abuf[i].y = Arow[(kk) + 4 * i + 1];                               \
      _Pragma("unroll")                                                 \
      for (int j = 0; j < NB; ++j) {                                    \
        if (nok[j]) {                                                   \
          bbuf[i][j].x = Wrow[j][(kk) + 4 * i];                         \
          bbuf[i][j].y = Wrow[j][(kk) + 4 * i + 1];                     \
        } else {                                                        \
          bbuf[i][j].x = 0.f; bbuf[i][j].y = 0.f;                       \
        }                                                               \
      }                                                                 \
    }                                                                   \
  } while (0)

#define MMA_CHUNK(abuf, bbuf)                                           \
  do {                                                                  \
    _Pragma("unroll")                                                   \
    for (int i = 0; i < 4; ++i) {                                       \
      _Pragma("unroll")                                                 \
      for (int j = 0; j < NB; ++j) {                                    \
        acc[j] = __builtin_amdgcn_wmma_f32_16x16x4_f32(                 \
            false, abuf[i], false, bbuf[i][j], (short)0, acc[j],        \
            false, false);                                              \
      }                                                                 \
    }                                                                   \
  } while (0)

  LOAD_CHUNK(a0, b0, 0);
  for (int k = 0; k + 32 <= K; k += 32) {
    LOAD_CHUNK(a1, b1, k + 16);                   // prefetch while mma on buf0
    MMA_CHUNK(a0, b0);
    if (k + 32 < K) LOAD_CHUNK(a0, b0, k + 32);   // prefetch while mma on buf1
    MMA_CHUNK(a1, b1);
  }

#undef LOAD_CHUNK
#undef MMA_CHUNK

#pragma unroll
  for (int j = 0; j < NB; ++j) {
    const int n = n0 + j * 16 + r15;
    if (NGUARD && n >= Nreal) continue;           // store-side guard only
    const float bi = bias ? bias[n] : 0.f;
#pragma unroll
    for (int r = 0; r < 8; ++r) {
      const int m = m0 + r + ((lane >> 4) << 3);
      float v = acc[j][r] + bi;
      if (ACCUM) v += C[(size_t)m * ldc + n];
      C[(size_t)m * ldc + n] = v;
    }
  }
}

// ---------------------------------------------------------------------------
// e[b,s] = sum_h w_score[h] * tanh(Hproj[b,s,h] + hp[b,h])
// one wave per (b,s) row; wave32 shuffle reduction
// ---------------------------------------------------------------------------
__global__ __launch_bounds__(256) void attn_energy(
    const float* __restrict__ Hproj, const float* __restrict__ hp,
    const float* __restrict__ wscore, float* __restrict__ e)
{
  const int lane = threadIdx.x & 31;
  const int wave = threadIdx.x >> 5;
  const int row  = blockIdx.x * 8 + wave;          // [0, B*S)
  const int b    = row >> 8;                       // S = 256
  const float* __restrict__ hrow = Hproj + (size_t)row * Hc;
  const float* __restrict__ hpb  = hp + (size_t)b * Hc;
  float acc = 0.f;
#pragma unroll
  for (int h = lane; h < Hc; h += 32)
    acc = fmaf(wscore[h], tanhf(hrow[h] + hpb[h]), acc);
#pragma unroll
  for (int off = 16; off > 0; off >>= 1)
    acc += __shfl_xor(acc, off, 32);
  if (lane == 0) e[row] = acc;
}

// ---------------------------------------------------------------------------
// in-place softmax over S=256 per batch row (one block per b)
// ---------------------------------------------------------------------------
__global__ __launch_bounds__(256) void softmax_rows(float* __restrict__ e)
{
  __shared__ float red[256];
  const int b = blockIdx.x;
  const int s = threadIdx.x;
  const float v = e[b * Sc + s];
  red[s] = v;
  __syncthreads();
#pragma unroll
  for (int off = 128; off > 0; off >>= 1) {
    if (s < off) red[s] = fmaxf(red[s], red[s + off]);
    __syncthreads();
  }
  const float m = red[0];
  __syncthreads();
  const float ex = __expf(v - m);
  red[s] = ex;
  __syncthreads();
#pragma unroll
  for (int off = 128; off > 0; off >>= 1) {
    if (s < off) red[s] += red[s + off];
    __syncthreads();
  }
  e[b * Sc + s] = ex / red[0];
}

// ---------------------------------------------------------------------------
// ctx[b,d] = sum_s alpha[b,s] * CF[b,s,d]   -> written into x[b, 0:D]
// grid = (D/256, B)
// ---------------------------------------------------------------------------
__global__ __launch_bounds__(256) void attn_ctx(
    const float* __restrict__ alpha, const float* __restrict__ CF,
    float* __restrict__ x)
{
  __shared__ float al[Sc];
  const int b = blockIdx.y;
  const int d = blockIdx.x * 256 + threadIdx.x;
  al[threadIdx.x] = alpha[b * Sc + threadIdx.x];
  __syncthreads();
  const float* __restrict__ cfb = CF + ((size_t)b * Sc) * Dc + d;
  float acc = 0.f;
#pragma unroll 4
  for (int s = 0; s < Sc; ++s)
    acc = fmaf(al[s], cfb[(size_t)s * Dc], acc);
  x[(size_t)b * DEc + d] = acc;
}

// ---------------------------------------------------------------------------
// embedding gather -> x[b, D : D+E];  tok = SOS at t==0 else labels[t-1,b]
// ---------------------------------------------------------------------------
__global__ __launch_bounds__(256) void embed_tok(
    const float* __restrict__ table, const int* __restrict__ labels,
    int t, float* __restrict__ x)
{
  const int b = blockIdx.x;
  const int e = threadIdx.x;
  const int tok = (t == 0) ? SOSc : labels[(t - 1) * Bc + b];
  x[(size_t)b * DEc + Dc + e] = table[(size_t)tok * Ec + e];
}

// ---------------------------------------------------------------------------
// LSTM pointwise update (gate order i,f,g,o), in-place h/c
// ---------------------------------------------------------------------------
__device__ __forceinline__ float sigmf(float v) { return 1.f / (1.f + __expf(-v)); }

__global__ __launch_bounds__(256) void lstm_point(
    const float* __restrict__ gates, float* __restrict__ h, float* __restrict__ c)
{
  const int i = blockIdx.x * 256 + threadIdx.x;    // [0, B*H)
  const int b = i >> 9;                            // H = 512
  const int hh = i & (Hc - 1);
  const float* __restrict__ g = gates + (size_t)b * G4c;
  const float ig = sigmf(g[hh]);
  const float fg = sigmf(g[Hc + hh]);
  const float gg = tanhf(g[2 * Hc + hh]);
  const float og = sigmf(g[3 * Hc + hh]);
  const float cn = fg * c[i] + ig * gg;
  c[i] = cn;
  h[i] = og * tanhf(cn);
}

// ---------------------------------------------------------------------------
// host driver
// ---------------------------------------------------------------------------
static inline int gemm_blocks(int tilesM, int tilesN) {
  return (tilesM * tilesN + 7) / 8;
}

extern "C" void kernel_launch(void* const* d_in, const int* in_sizes, int n_in,
                              void* d_out, int out_size, void* d_ws, size_t ws_size,
                              hipStream_t stream)
{
  const float* CF      = (const float*)d_in[0];
  const float* h0      = (const float*)d_in[1];
  const float* c0      = (const float*)d_in[2];
  const int*   labels  = (const int*)d_in[3];
  const float* table   = (const float*)d_in[5];
  const float* W_i2h   = (const float*)d_in[6];
  const float* W_h2h   = (const float*)d_in[7];
  const float* b_h2h   = (const float*)d_in[8];
  const float* w_score = (const float*)d_in[9];
  const float* W_ih    = (const float*)d_in[10];
  const float* W_hh    = (const float*)d_in[11];
  const float* b_ih    = (const float*)d_in[12];
  const float* b_hh    = (const float*)d_in[13];
  const float* W_gen   = (const float*)d_in[14];
  const float* b_gen   = (const float*)d_in[15];
  float* out = (float*)d_out;

  // workspace carve-up (floats)
  float* ws    = (float*)d_ws;
  float* Hproj = ws;                                  // B*S*H
  float* hbuf  = Hproj + (size_t)Bc * Sc * Hc;        // B*H
  float* cbuf  = hbuf + (size_t)Bc * Hc;              // B*H
  float* hp    = cbuf + (size_t)Bc * Hc;              // B*H
  float* ebuf  = hp   + (size_t)Bc * Hc;              // B*S
  float* xbuf  = ebuf + (size_t)Bc * Sc;              // B*(D+E)
  float* gates = xbuf + (size_t)Bc * DEc;             // B*4H

  const int T = in_sizes[3] / Bc;                     // labels are [T,B]

  // init recurrent state
  hipMemcpyAsync(hbuf, h0, sizeof(float) * Bc * Hc, hipMemcpyDeviceToDevice, stream);
  hipMemcpyAsync(cbuf, c0, sizeof(float) * Bc * Hc, hipMemcpyDeviceToDevice, stream);

  // one-time: Hproj[B*S, H] = CF[B*S, D] @ W_i2h[H, D]^T
  {
    const int tm = (Bc * Sc) / 16, tn = Hc / 64;
    gemm_nt_wmma<4, false, false><<<gemm_blocks(tm, tn), 256, 0, stream>>>(
        CF, W_i2h, nullptr, Hproj, Bc * Sc, Hc, Dc, Hc, Hc);
  }

  for (int t = 0; t < T; ++t) {
    // hp = h @ W_h2h^T + b_h2h
    gemm_nt_wmma<4, false, false><<<gemm_blocks(Bc / 16, Hc / 64), 256, 0, stream>>>(
        hbuf, W_h2h, b_h2h, hp, Bc, Hc, Hc, Hc, Hc);

    // e[b,s] = w_score . tanh(Hproj + hp)
    attn_energy<<<(Bc * Sc) / 8, 256, 0, stream>>>(Hproj, hp, w_score, ebuf);

    // alpha = softmax(e)  (in place)
    softmax_rows<<<Bc, 256, 0, stream>>>(ebuf);

    // x[:, 0:D] = alpha @ CF
    attn_ctx<<<dim3(Dc / 256, Bc), 256, 0, stream>>>(ebuf, CF, xbuf);

    // x[:, D:D+E] = embedding[tok]
    embed_tok<<<Bc, 256, 0, stream>>>(table, labels, t, xbuf);

    // gates = x @ W_ih^T + b_ih
    gemm_nt_wmma<4, false, false><<<gemm_blocks(Bc / 16, G4c / 64), 256, 0, stream>>>(
        xbuf, W_ih, b_ih, gates, Bc, G4c, DEc, G4c, G4c);
    // gates += h @ W_hh^T + b_hh
    gemm_nt_wmma<4, true, false><<<gemm_blocks(Bc / 16, G4c / 64), 256, 0, stream>>>(
        hbuf, W_hh, b_hh, gates, Bc, G4c, Hc, G4c, G4c);

    // LSTM pointwise: updates h, c in place
    lstm_point<<<(Bc * Hc) / 256, 256, 0, stream>>>(gates, hbuf, cbuf);

    // logits[t] = h @ W_gen^T + b_gen   (N = 101, padded to 112 with guards)
    gemm_nt_wmma<1, false, true><<<gemm_blocks(Bc / 16, 112 / 16), 256, 0, stream>>>(
        hbuf, W_gen, b_gen, out + (size_t)t * Bc * OUTc, Bc, 112, Hc, OUTc, OUTc);
  }
}